// phase_transforms_24653112279144
// MI455X (gfx1250) — compile-verified
//
#include <hip/hip_runtime.h>
#include <stdint.h>

// Elementwise complex phase rotation:
//   out_real = x_real*cos(p) - x_imag*sin(p)
//   out_imag = x_real*sin(p) + x_imag*cos(p)
// Memory-bound: 4 x 128 MiB streams = 512 MiB HBM traffic, ~23 us floor at
// 23.3 TB/s. No matrix op exists here (diag matmul == elementwise), so the
// CDNA5 wins are data movement: 128-bit NT loads/stores + async global->LDS
// staging of the cos/sin table (ASYNCcnt path), and high occupancy.

typedef float v4f __attribute__((ext_vector_type(4)));

#ifndef PHASE_MAX_NDIM
#define PHASE_MAX_NDIM 2048   // reference NDIM; LDS table sized for this
#endif

// ---------------------------------------------------------------------------
// Kernel 1: fill ws[0..ndim) = cos(phases), ws[ndim..2*ndim) = sin(phases).
// Trivial cost (ndim threads); precise libm sin/cos lives ONLY here so the
// streaming kernel stays register-lean.
// ---------------------------------------------------------------------------
__global__ void phase_table_kernel(const float* __restrict__ phases,
                                   float* __restrict__ ws, int ndim) {
  int i = blockIdx.x * blockDim.x + threadIdx.x;
  if (i < ndim) {
    float p = phases[i];
    ws[i]        = cosf(p);
    ws[ndim + i] = sinf(p);
  }
}

// ---------------------------------------------------------------------------
// Shared streaming body: assumes cs[] (cos|sin, 2*ndim floats) is populated
// and the block barrier has been passed.
// ---------------------------------------------------------------------------
__device__ __forceinline__
void stream_rows(const float* __restrict__ xr, const float* __restrict__ xi,
                 float* __restrict__ out, const float* cs,
                 int Brows, int ndim, int rowsPerBlock) {
  const int t   = threadIdx.x;
  const int nd4 = ndim >> 2;                      // float4s per row
  const long long tot4 = (long long)Brows * nd4;  // float4s per plane
  int r0 = blockIdx.x * rowsPerBlock;
  int r1 = r0 + rowsPerBlock;
  if (r1 > Brows) r1 = Brows;

  const v4f* __restrict__ xr4 = (const v4f*)xr;
  const v4f* __restrict__ xi4 = (const v4f*)xi;
  v4f* __restrict__ or4 = (v4f*)out;
  v4f* __restrict__ oi4 = (v4f*)out + tot4;       // imag plane follows real

  if (nd4 == 2 * (int)blockDim.x) {
    // Specialized common case (NDIM=2048, 256 threads): each thread owns two
    // fixed float4 column slots; c/s live in VGPRs across the whole row loop,
    // and each row iteration issues 4 independent NT loads up front (MLP).
    const int j0 = t, j1 = t + (int)blockDim.x;
    const v4f c0 = *(const v4f*)&cs[4 * j0];
    const v4f s0 = *(const v4f*)&cs[ndim + 4 * j0];
    const v4f c1 = *(const v4f*)&cs[4 * j1];
    const v4f s1 = *(const v4f*)&cs[ndim + 4 * j1];
    for (int r = r0; r < r1; ++r) {
      long long i0 = (long long)r * nd4 + j0;
      long long i1 = (long long)r * nd4 + j1;
      v4f a0 = __builtin_nontemporal_load(xr4 + i0);
      v4f b0 = __builtin_nontemporal_load(xi4 + i0);
      v4f a1 = __builtin_nontemporal_load(xr4 + i1);
      v4f b1 = __builtin_nontemporal_load(xi4 + i1);
      v4f re0 = a0 * c0 - b0 * s0;
      v4f im0 = a0 * s0 + b0 * c0;
      v4f re1 = a1 * c1 - b1 * s1;
      v4f im1 = a1 * s1 + b1 * c1;
      __builtin_nontemporal_store(re0, or4 + i0);
      __builtin_nontemporal_store(im0, oi4 + i0);
      __builtin_nontemporal_store(re1, or4 + i1);
      __builtin_nontemporal_store(im1, oi4 + i1);
    }
  } else {
    for (int j = t; j < nd4; j += (int)blockDim.x) {
      v4f c = *(const v4f*)&cs[4 * j];
      v4f s = *(const v4f*)&cs[ndim + 4 * j];
      for (int r = r0; r < r1; ++r) {
        long long idx = (long long)r * nd4 + j;
        v4f a = __builtin_nontemporal_load(xr4 + idx);
        v4f b = __builtin_nontemporal_load(xi4 + idx);
        __builtin_nontemporal_store(a * c - b * s, or4 + idx);
        __builtin_nontemporal_store(a * s + b * c, oi4 + idx);
      }
    }
  }
}

// ---------------------------------------------------------------------------
// Kernel 2 (hot path): async-stage the 16 KB cos/sin table from ws into LDS
// with gfx1250 global_load_async_to_lds_b128 (tracked by ASYNCcnt), then
// stream. No libm code in this kernel -> lean VGPR budget -> max occupancy.
// ---------------------------------------------------------------------------
__global__ __launch_bounds__(256)
void phase_rot_kernel(const float* __restrict__ xr,
                      const float* __restrict__ xi,
                      const float* __restrict__ ws,
                      float* __restrict__ out,
                      int Brows, int ndim, int rowsPerBlock) {
  __shared__ __attribute__((aligned(16))) float cs[2 * PHASE_MAX_NDIM];

  // Generic pointer to __shared__ carries the wave-relative LDS offset in its
  // low 32 bits (flat->LDS mapping truncates to addr[31:0]).
  unsigned ldsbase = (unsigned)(unsigned long long)(uintptr_t)(void*)cs;
  const char* gbase = (const char*)ws;
  const int n16 = (2 * ndim) >> 2;  // 16-byte chunks (1024 for ndim=2048)
  for (int e = threadIdx.x; e < n16; e += (int)blockDim.x) {
    unsigned loff = ldsbase + ((unsigned)e << 4);
    unsigned long long gaddr = (unsigned long long)(uintptr_t)(gbase + ((size_t)e << 4));
    asm volatile("global_load_async_to_lds_b128 %0, %1, off"
                 :: "v"(loff), "v"(gaddr)
                 : "memory");
  }
  asm volatile("s_wait_asynccnt 0" ::: "memory");
  __syncthreads();

  stream_rows(xr, xi, out, cs, Brows, ndim, rowsPerBlock);
}

// ---------------------------------------------------------------------------
// Kernel 2 fallback (only if no workspace): compute the table in-block.
// Keeps the libm bloat out of the hot kernel.
// ---------------------------------------------------------------------------
__global__ __launch_bounds__(256)
void phase_rot_nows_kernel(const float* __restrict__ xr,
                           const float* __restrict__ xi,
                           const float* __restrict__ phases,
                           float* __restrict__ out,
                           int Brows, int ndim, int rowsPerBlock) {
  __shared__ __attribute__((aligned(16))) float cs[2 * PHASE_MAX_NDIM];
  for (int i = threadIdx.x; i < ndim; i += (int)blockDim.x) {
    float p = phases[i];
    cs[i]        = cosf(p);
    cs[ndim + i] = sinf(p);
  }
  __syncthreads();
  stream_rows(xr, xi, out, cs, Brows, ndim, rowsPerBlock);
}

// ---------------------------------------------------------------------------
// Host launcher (graph-capture safe: only kernel launches on `stream`).
// ---------------------------------------------------------------------------
extern "C" void kernel_launch(void* const* d_in, const int* in_sizes, int n_in,
                              void* d_out, int out_size, void* d_ws, size_t ws_size,
                              hipStream_t stream) {
  const float* xr = (const float*)d_in[0];   // [B, NDIM] f32
  const float* xi = (const float*)d_in[1];   // [B, NDIM] f32
  const float* ph = (const float*)d_in[2];   // [NDIM]    f32
  float* out = (float*)d_out;                // [2, B, NDIM] f32

  const int ndim  = in_sizes[2];
  const int Brows = in_sizes[0] / ndim;

  const int rowsPerBlock = 8;  // 16 KB staging amortized over 256 KB payload
  const int grid = (Brows + rowsPerBlock - 1) / rowsPerBlock;  // 2048 blocks

  const size_t table_bytes = (size_t)(2 * ndim) * sizeof(float);
  if (ws_size >= table_bytes && ndim <= PHASE_MAX_NDIM) {
    float* ws = (float*)d_ws;
    phase_table_kernel<<<(ndim + 255) / 256, 256, 0, stream>>>(ph, ws, ndim);
    phase_rot_kernel<<<grid, 256, 0, stream>>>(xr, xi, ws, out,
                                               Brows, ndim, rowsPerBlock);
  } else {
    phase_rot_nows_kernel<<<grid, 256, 0, stream>>>(xr, xi, ph, out,
                                                    Brows, ndim, rowsPerBlock);
  }
}